// SimplePoseGNN_21371757265053
// MI455X (gfx1250) — compile-verified
//
#include <hip/hip_runtime.h>

#define NN   100000     // nodes  (== 6250 * 16, multiple of 16)
#define NE   1600000    // edges
#define NG   2000       // graphs
#define IND  6
#define HID  256
#define ODIM 128
#define NCLS 60
#define EPSV 1e-5f

typedef __attribute__((ext_vector_type(2))) float v2f;
typedef __attribute__((ext_vector_type(8))) float v8f;

// ---------------------------------------------------------------------------
// Degrees: one thread per edge, fire-and-forget f32 atomics
// ---------------------------------------------------------------------------
__global__ __launch_bounds__(256)
void degree_kernel(const int* __restrict__ src, const int* __restrict__ dst,
                   float* __restrict__ dout, float* __restrict__ din) {
    int e = blockIdx.x * blockDim.x + threadIdx.x;
    if (e >= NE) return;
    unsafeAtomicAdd(&dout[src[e]], 1.0f);
    unsafeAtomicAdd(&din[dst[e]], 1.0f);
}

__global__ __launch_bounds__(256)
void isqrt_kernel(float* __restrict__ d, int n) {
    int i = blockIdx.x * blockDim.x + threadIdx.x;
    if (i >= n) return;
    float v = d[i];
    d[i] = (v > 0.0f) ? __frsqrt_rn(fmaxf(v, 1.0f)) : 0.0f;
}

// ---------------------------------------------------------------------------
// Embedding: X = node_features @ W_emb + b_emb   (K=6, scalar)
// ---------------------------------------------------------------------------
__global__ __launch_bounds__(256)
void embed_kernel(const float* __restrict__ nf, const float* __restrict__ W,
                  const float* __restrict__ b, float* __restrict__ X) {
    int idx = blockIdx.x * blockDim.x + threadIdx.x;
    if (idx >= NN * HID) return;
    int n = idx / HID, c = idx - n * HID;
    float acc = b[c];
#pragma unroll
    for (int i = 0; i < IND; ++i) acc += nf[n * IND + i] * W[i * HID + c];
    X[idx] = acc;
}

// ---------------------------------------------------------------------------
// Weight pre-pack: Wp[(kk/2)*NC + col][kk&1] = W[kk][col]
// -> a lane's B fragment {W[kk][col], W[kk+1][col]} becomes one b64 load
// ---------------------------------------------------------------------------
__global__ __launch_bounds__(256)
void pack_w_kernel(const float* __restrict__ W, float* __restrict__ Wp, int K, int NC) {
    int idx = blockIdx.x * blockDim.x + threadIdx.x;
    if (idx >= K * NC) return;
    int kk = idx / NC, col = idx - kk * NC;
    Wp[(size_t)((kk >> 1) * NC + col) * 2 + (kk & 1)] = W[idx];
}

// ---------------------------------------------------------------------------
// Edge scatter: agg[dst] += x[src] * dout_isqrt[src]
// ---------------------------------------------------------------------------
__global__ __launch_bounds__(64)
void scatter_kernel(const float* __restrict__ Xsrc, const int* __restrict__ src,
                    const int* __restrict__ dst, const float* __restrict__ dscale,
                    float* __restrict__ agg) {
    int e = blockIdx.x;
    int s = src[e], d = dst[e];
    float sc = dscale[s];
    const float4* xs = (const float4*)(Xsrc + (size_t)s * HID);
    float4 v = xs[threadIdx.x];
    float* out = agg + (size_t)d * HID + threadIdx.x * 4;
    unsafeAtomicAdd(out + 0, v.x * sc);
    unsafeAtomicAdd(out + 1, v.y * sc);
    unsafeAtomicAdd(out + 2, v.z * sc);
    unsafeAtomicAdd(out + 3, v.w * sc);
}

// ---------------------------------------------------------------------------
// FP32 WMMA GEMM: Y[NN,NC] = diag(rowscale) * A[NN,K] @ W[K,NC] + bias (+res)
// K = HID (compile-time). One wave computes a 16x64 output strip (4 tiles),
// reusing the A fragment across 4 column tiles; software-pipelined loads.
// A frag (16x4, ISA 7.12.2): lanes 0-15 hold K=0..1, lanes 16-31 K=2..3.
// D frag: VGPR i -> row i (lanes 0-15) / row 8+i (lanes 16-31), col = lane%16.
// ---------------------------------------------------------------------------
template <int NC, bool RESID, bool RSCALE>
__global__ __launch_bounds__(256)
void wmma_gemm_f32(const float* __restrict__ A, const float* __restrict__ Wp,
                   const float* __restrict__ bias, const float* __restrict__ rowscale,
                   const float* __restrict__ residual, float* __restrict__ Y) {
    constexpr int K  = HID;
    constexpr int TJ = NC / 64;                       // column tiles per wave
    const int wid = blockIdx.x * 8 + (threadIdx.x >> 5);
    if (wid >= (NN / 16) * TJ) return;                // uniform per wave
    const int tm = wid / TJ;
    const int tj = wid - tm * TJ;
    const int lane = threadIdx.x & 31;
    const int half = lane >> 4;                       // K-pair select
    const int l16  = lane & 15;
    const int row     = tm * 16 + l16;                // A row for this lane
    const int colbase = tj * 64 + l16;                // first B/D column

    const float rs = RSCALE ? rowscale[row] : 1.0f;
    const float* Ap = A + (size_t)row * K + 2 * half;
    const float* Bp = Wp + ((size_t)half * NC + colbase) * 2;

    v8f acc[TJ];
#pragma unroll
    for (int j = 0; j < TJ; ++j) acc[j] = (v8f){};

    // prologue: fragments for k = 0
    v2f a = *(const v2f*)Ap;
    v2f b[TJ];
#pragma unroll
    for (int j = 0; j < TJ; ++j) b[j] = *(const v2f*)(Bp + 32 * j);

#pragma unroll 4
    for (int k = 0; k < K - 4; k += 4) {
        // prefetch k+4 while WMMAs for k issue
        v2f an = *(const v2f*)(Ap + k + 4);
        v2f bn[TJ];
#pragma unroll
        for (int j = 0; j < TJ; ++j)
            bn[j] = *(const v2f*)(Bp + (k + 4) * NC + 32 * j);

        v2f as; as.x = a.x * rs; as.y = a.y * rs;
#pragma unroll
        for (int j = 0; j < TJ; ++j)
            acc[j] = __builtin_amdgcn_wmma_f32_16x16x4_f32(
                false, as, false, b[j], (short)0, acc[j], false, false);

        a = an;
#pragma unroll
        for (int j = 0; j < TJ; ++j) b[j] = bn[j];
    }
    {   // final k-step
        v2f as; as.x = a.x * rs; as.y = a.y * rs;
#pragma unroll
        for (int j = 0; j < TJ; ++j)
            acc[j] = __builtin_amdgcn_wmma_f32_16x16x4_f32(
                false, as, false, b[j], (short)0, acc[j], false, false);
    }

    float bv[TJ];
#pragma unroll
    for (int j = 0; j < TJ; ++j) bv[j] = bias[colbase + 16 * j];

#pragma unroll
    for (int i = 0; i < 8; ++i) {
        const int r = tm * 16 + i + 8 * half;
        const size_t o = (size_t)r * NC + colbase;
#pragma unroll
        for (int j = 0; j < TJ; ++j) {
            float v = acc[j][i] + bv[j];
            if (RESID) v += residual[o + 16 * j];
            Y[o + 16 * j] = v;
        }
    }
}

// ---------------------------------------------------------------------------
// BatchNorm: per-channel sum / sumsq, then fused normalize + ReLU
// ---------------------------------------------------------------------------
__global__ __launch_bounds__(HID)
void bn_stats_kernel(const float* __restrict__ H, float* __restrict__ stats,
                     int rows_per_block) {
    int c = threadIdx.x;
    int r0 = blockIdx.x * rows_per_block;
    int r1 = min(r0 + rows_per_block, NN);
    float s = 0.0f, ss = 0.0f;
    for (int r = r0; r < r1; ++r) {
        float v = H[(size_t)r * HID + c];
        s += v; ss += v * v;
    }
    unsafeAtomicAdd(&stats[c], s);
    unsafeAtomicAdd(&stats[HID + c], ss);
}

__global__ __launch_bounds__(256)
void bn_apply_relu_kernel(float* __restrict__ H, const float* __restrict__ stats,
                          const float* __restrict__ g, const float* __restrict__ b) {
    int idx = blockIdx.x * blockDim.x + threadIdx.x;
    if (idx >= NN * HID) return;
    int c = idx & (HID - 1);
    const float invn = 1.0f / (float)NN;
    float mu  = stats[c] * invn;
    float var = stats[HID + c] * invn - mu * mu;
    float v = (H[idx] - mu) * __frsqrt_rn(var + EPSV) * g[c] + b[c];
    H[idx] = v > 0.0f ? v : 0.0f;
}

// ---------------------------------------------------------------------------
// Graph mean-pool + classifier
// ---------------------------------------------------------------------------
__global__ __launch_bounds__(ODIM)
void pool_kernel(const float* __restrict__ Hout, const int* __restrict__ gid,
                 float* __restrict__ pool, float* __restrict__ cnt) {
    int n = blockIdx.x;
    int g = gid[n];
    int c = threadIdx.x;
    unsafeAtomicAdd(&pool[(size_t)g * ODIM + c], Hout[(size_t)n * ODIM + c]);
    if (c == 0) unsafeAtomicAdd(&cnt[g], 1.0f);
}

__global__ __launch_bounds__(256)
void cls_kernel(const float* __restrict__ pool, const float* __restrict__ cnt,
                const float* __restrict__ Wc, const float* __restrict__ bc,
                float* __restrict__ label) {
    int idx = blockIdx.x * blockDim.x + threadIdx.x;
    if (idx >= NG * NCLS) return;
    int g = idx / NCLS, c = idx - g * NCLS;
    float inv = 1.0f / fmaxf(cnt[g], 1.0f);
    float acc = bc[c];
    for (int k = 0; k < ODIM; ++k)
        acc += pool[(size_t)g * ODIM + k] * inv * Wc[k * NCLS + c];
    label[idx] = acc;
}

// ---------------------------------------------------------------------------
extern "C" void kernel_launch(void* const* d_in, const int* in_sizes, int n_in,
                              void* d_out, int out_size, void* d_ws, size_t ws_size,
                              hipStream_t stream) {
    const float* nf   = (const float*)d_in[0];
    const int*   src  = (const int*)  d_in[1];
    const int*   dst  = (const int*)  d_in[2];
    const int*   gid  = (const int*)  d_in[3];
    const float* Wemb = (const float*)d_in[4];
    const float* bemb = (const float*)d_in[5];
    const float* Wc1  = (const float*)d_in[6];
    const float* bc1  = (const float*)d_in[7];
    const float* g1   = (const float*)d_in[8];
    const float* be1  = (const float*)d_in[9];
    const float* W2   = (const float*)d_in[10];
    const float* b2   = (const float*)d_in[11];
    const float* Wc2  = (const float*)d_in[12];
    const float* bc2  = (const float*)d_in[13];
    const float* g2   = (const float*)d_in[14];
    const float* be2  = (const float*)d_in[15];
    const float* W3   = (const float*)d_in[16];
    const float* b3   = (const float*)d_in[17];
    const float* Wcls = (const float*)d_in[18];
    const float* bcls = (const float*)d_in[19];

    float* out_h     = (float*)d_out;                       // [NN, ODIM]
    float* out_label = out_h + (size_t)NN * ODIM;           // [NG, NCLS]

    const size_t NH = (size_t)NN * HID;
    float* X      = (float*)d_ws;         // embedding (kept for residuals)
    float* AGG    = X + NH;               // scatter-add target
    float* HA     = AGG + NH;             // working activation
    float* HB     = HA + NH;              // working activation
    float* dout_d = HB + NH;              // out-degree -> dout^-1/2 (in place)
    float* din_d  = dout_d + NN;          // in-degree  -> din^-1/2  (in place)
    float* stats  = din_d + NN;           // [2*HID] sum / sumsq
    float* pool   = stats + 2 * HID;      // [NG, ODIM]
    float* cnt    = pool + (size_t)NG * ODIM;
    float* Wc1p   = cnt + NG;             // packed weights
    float* W2p    = Wc1p + HID * HID;
    float* Wc2p   = W2p  + HID * HID;
    float* W3p    = Wc2p + HID * HID;

    // --- pack weights for b64 B-fragment loads ---
    pack_w_kernel<<<(HID * HID + 255) / 256, 256, 0, stream>>>(Wc1, Wc1p, HID, HID);
    pack_w_kernel<<<(HID * HID + 255) / 256, 256, 0, stream>>>(W2,  W2p,  HID, HID);
    pack_w_kernel<<<(HID * HID + 255) / 256, 256, 0, stream>>>(Wc2, Wc2p, HID, HID);
    pack_w_kernel<<<(HID * ODIM + 255) / 256, 256, 0, stream>>>(W3, W3p, HID, ODIM);

    // --- degrees -> isqrt (dout_d/din_d are contiguous) ---
    hipMemsetAsync(dout_d, 0, 2 * NN * sizeof(float), stream);
    degree_kernel<<<(NE + 255) / 256, 256, 0, stream>>>(src, dst, dout_d, din_d);
    isqrt_kernel<<<(2 * NN + 255) / 256, 256, 0, stream>>>(dout_d, 2 * NN);

    // --- embedding ---
    embed_kernel<<<(NN * HID + 255) / 256, 256, 0, stream>>>(nf, Wemb, bemb, X);

    const int blk_hh = ((NN / 16) * (HID / 64) + 7) / 8;   // 16x64 strips, 8 waves/block
    const int blk_ho = ((NN / 16) * (ODIM / 64) + 7) / 8;

    // --- conv1: agg = scatter(x * dout^-1/2); HA = diag(din^-1/2)*agg @ Wc1 + bc1 + X
    hipMemsetAsync(AGG, 0, NH * sizeof(float), stream);
    scatter_kernel<<<NE, 64, 0, stream>>>(X, src, dst, dout_d, AGG);
    wmma_gemm_f32<HID, true, true><<<blk_hh, 256, 0, stream>>>(AGG, Wc1p, bc1, din_d, X, HA);

    // --- bn1 + relu ---
    hipMemsetAsync(stats, 0, 2 * HID * sizeof(float), stream);
    bn_stats_kernel<<<(NN + 127) / 128, HID, 0, stream>>>(HA, stats, 128);
    bn_apply_relu_kernel<<<(NN * HID + 255) / 256, 256, 0, stream>>>(HA, stats, g1, be1);

    // --- HB = HA @ W2 + b2 ---
    wmma_gemm_f32<HID, false, false><<<blk_hh, 256, 0, stream>>>(HA, W2p, b2, nullptr, nullptr, HB);

    // --- conv2: agg = scatter(HB * dout^-1/2); HA = diag(din^-1/2)*agg @ Wc2 + bc2 + X
    hipMemsetAsync(AGG, 0, NH * sizeof(float), stream);
    scatter_kernel<<<NE, 64, 0, stream>>>(HB, src, dst, dout_d, AGG);
    wmma_gemm_f32<HID, true, true><<<blk_hh, 256, 0, stream>>>(AGG, Wc2p, bc2, din_d, X, HA);

    // --- bn2 + relu ---
    hipMemsetAsync(stats, 0, 2 * HID * sizeof(float), stream);
    bn_stats_kernel<<<(NN + 127) / 128, HID, 0, stream>>>(HA, stats, 128);
    bn_apply_relu_kernel<<<(NN * HID + 255) / 256, 256, 0, stream>>>(HA, stats, g2, be2);

    // --- out_h = HA @ W3 + b3  (ODIM columns) ---
    wmma_gemm_f32<ODIM, false, false><<<blk_ho, 256, 0, stream>>>(HA, W3p, b3, nullptr, nullptr, out_h);

    // --- per-graph mean pool + classifier ---
    hipMemsetAsync(pool, 0, ((size_t)NG * ODIM + NG) * sizeof(float), stream);
    pool_kernel<<<NN, ODIM, 0, stream>>>(out_h, gid, pool, cnt);
    cls_kernel<<<(NG * NCLS + 255) / 256, 256, 0, stream>>>(pool, cnt, Wcls, bcls, out_label);
}